// SparseAutoencoder_Conv_3702261809451
// MI455X (gfx1250) — compile-verified
//
#include <hip/hip_runtime.h>
#include <math.h>

#define CHW        98304      // C*H*W = 1*128*768
#define BATCH      512
#define NUM_CLS    5
#define TOPK       64
#define INV_TAU    10.0f

typedef __attribute__((ext_vector_type(2))) float v2f;
typedef __attribute__((ext_vector_type(4))) float v4f;
typedef __attribute__((ext_vector_type(8))) float v8f;

// ---------------------------------------------------------------------------
// Kernel 1: single workgroup. Computes gumbel scores, softmax stats (max, sum),
// iterative top-64 argmax, and the final mask vector (exact zeros except the
// 64 winners, where value = (1 - soft) + soft, matching the reference fp32).
// ---------------------------------------------------------------------------
__global__ void __launch_bounds__(1024)
mask_topk_kernel(const float* __restrict__ logits,
                 const float* __restrict__ gumbel,
                 float* __restrict__ scores,   // [CHW] scratch
                 float* __restrict__ mask,     // [CHW] final mask
                 int*   __restrict__ topk,     // [64]
                 float* __restrict__ fmv)      // [64] mask values at winners
{
    __shared__ float red_f[1024];
    __shared__ int   red_i[1024];
    __shared__ float s_M, s_sum;

    const int tid = threadIdx.x;

    // ---- Phase A: scores, mask init, local max of logits ----
    float lmax = -__builtin_inff();
    for (int i = tid; i < CHW; i += 1024) {
        float l = logits[i];
        scores[i] = (l + gumbel[i]) * INV_TAU;
        mask[i]   = 0.0f;               // (0 - s) + s == +0 exactly
        lmax = fmaxf(lmax, l);
    }
    red_f[tid] = lmax;
    __syncthreads();
    for (int s = 512; s > 0; s >>= 1) {
        if (tid < s) red_f[tid] = fmaxf(red_f[tid], red_f[tid + s]);
        __syncthreads();
    }
    if (tid == 0) s_M = red_f[0] * INV_TAU;
    __syncthreads();
    const float M = s_M;

    // ---- Phase A2: sum of exp(l/tau - M) ----
    float lsum = 0.0f;
    for (int i = tid; i < CHW; i += 1024)
        lsum += expf(logits[i] * INV_TAU - M);
    red_f[tid] = lsum;
    __syncthreads();
    for (int s = 512; s > 0; s >>= 1) {
        if (tid < s) red_f[tid] += red_f[tid + s];
        __syncthreads();
    }
    if (tid == 0) s_sum = red_f[0];
    __syncthreads();
    const float denom = s_sum;

    // ---- Phase B: 64 iterations of block-wide argmax (first-occurrence ties) ----
    for (int it = 0; it < TOPK; ++it) {
        float bv = -__builtin_inff();
        int   bi = 0x7fffffff;
        for (int i = tid; i < CHW; i += 1024) {
            float v = scores[i];
            if (v > bv) { bv = v; bi = i; }   // strided+strict '>' keeps lowest index
        }
        red_f[tid] = bv; red_i[tid] = bi;
        __syncthreads();
        for (int s = 512; s > 0; s >>= 1) {
            if (tid < s) {
                float ov = red_f[tid + s]; int oi = red_i[tid + s];
                if (ov > red_f[tid] || (ov == red_f[tid] && oi < red_i[tid])) {
                    red_f[tid] = ov; red_i[tid] = oi;
                }
            }
            __syncthreads();
        }
        if (tid == 0) {
            int w = red_i[0];
            topk[it]  = w;
            scores[w] = -__builtin_inff();
            __threadfence_block();
        }
        __syncthreads();
    }

    // ---- Phase C: mask values at winners: (1 - soft) + soft ----
    if (tid < TOPK) {
        int   j  = topk[tid];
        float s  = expf(logits[j] * INV_TAU - M) / denom;
        float fm = (1.0f - s) + s;
        mask[j]  = fm;
        fmv[tid] = fm;
    }
}

// ---------------------------------------------------------------------------
// Kernel 2: broadcast mask [CHW] to binary_mask [B, CHW].  Pure store-bound
// (201 MB).  Each block loads its 256-float4 segment once and streams 64 rows
// with non-temporal b128 stores (output > L2 capacity).
// Grid: 8 row-groups x 96 column-blocks = 768 blocks of 256 threads.
// ---------------------------------------------------------------------------
__global__ void __launch_bounds__(256)
broadcast_mask_kernel(const float* __restrict__ mask,
                      float* __restrict__ out)
{
    const int f4_per_row = CHW / 4;               // 24576
    const int jb  = blockIdx.x % 96;              // column block
    const int rg  = blockIdx.x / 96;              // row group (0..7)
    const int j4  = jb * 256 + threadIdx.x;       // float4 index within row

    const v4f* m4 = reinterpret_cast<const v4f*>(mask);
    v4f v = m4[j4];

    v4f* o4 = reinterpret_cast<v4f*>(out);
    const int row0 = rg * 64;
    #pragma unroll 4
    for (int r = 0; r < 64; ++r) {
        size_t off = (size_t)(row0 + r) * f4_per_row + j4;
        __builtin_nontemporal_store(v, &o4[off]);
    }
}

// ---------------------------------------------------------------------------
// Kernel 3: logits[512,5] = gather(x)[512,64] * diag(fm) @ gather(W)[64,16] + b
// via v_wmma_f32_16x16x4_f32.  B-matrix (64x16, n>=5 zero-padded) built in LDS.
// A-fragment gathered from x with the mask value folded in.
// Grid: 4 blocks x 8 waves = 32 M-tiles of 16 rows.
// ---------------------------------------------------------------------------
__global__ void __launch_bounds__(256)
logits_wmma_kernel(const float* __restrict__ x,
                   const float* __restrict__ Wh,     // [5, CHW]
                   const float* __restrict__ bias,   // [5]
                   const int*   __restrict__ topk,   // [64]
                   const float* __restrict__ fmv,    // [64]
                   float* __restrict__ out_logits)   // [512, 5]
{
    __shared__ float sB[TOPK * 16];   // B[k][n]
    __shared__ int   sIdx[TOPK];
    __shared__ float sFm[TOPK];

    const int tid = threadIdx.x;
    if (tid < TOPK) { sIdx[tid] = topk[tid]; sFm[tid] = fmv[tid]; }
    __syncthreads();

    for (int e = tid; e < TOPK * 16; e += 256) {
        int k = e >> 4, n = e & 15;
        sB[e] = (n < NUM_CLS) ? Wh[(size_t)n * CHW + sIdx[k]] : 0.0f;
    }
    __syncthreads();

    const int wave = tid >> 5;          // 0..7
    const int lane = tid & 31;
    const int half = lane >> 4;         // 0: K+0/K+1,  1: K+2/K+3 (A & B layout)
    const int l15  = lane & 15;
    const int m0   = (blockIdx.x * 8 + wave) * 16;

    const float* xrow = x + (size_t)(m0 + l15) * CHW;   // A row = m0 + (lane&15)

    v8f c = {0.f, 0.f, 0.f, 0.f, 0.f, 0.f, 0.f, 0.f};
    #pragma unroll
    for (int kk = 0; kk < 16; ++kk) {
        const int k0 = kk * 4 + half * 2;
        v2f a, b;
        a.x = xrow[sIdx[k0]]     * sFm[k0];
        a.y = xrow[sIdx[k0 + 1]] * sFm[k0 + 1];
        b.x = sB[k0 * 16 + l15];
        b.y = sB[(k0 + 1) * 16 + l15];
        c = __builtin_amdgcn_wmma_f32_16x16x4_f32(
                /*neg_a=*/false, a, /*neg_b=*/false, b,
                /*c_mod=*/(short)0, c, /*reuse_a=*/false, /*reuse_b=*/false);
    }

    // D layout: VGPR v -> row m0 + half*8 + v, col = lane&15; keep cols < 5
    if (l15 < NUM_CLS) {
        const float bb = bias[l15];
        #pragma unroll
        for (int v = 0; v < 8; ++v) {
            int row = m0 + half * 8 + v;
            out_logits[(size_t)row * NUM_CLS + l15] = c[v] + bb;
        }
    }
}

// ---------------------------------------------------------------------------
extern "C" void kernel_launch(void* const* d_in, const int* in_sizes, int n_in,
                              void* d_out, int out_size, void* d_ws, size_t ws_size,
                              hipStream_t stream)
{
    const float* x    = (const float*)d_in[0];   // [512,1,128,768]
    const float* mlog = (const float*)d_in[1];   // [CHW]
    const float* gum  = (const float*)d_in[2];   // [CHW]
    const float* Wh   = (const float*)d_in[3];   // [5, CHW]
    const float* bh   = (const float*)d_in[4];   // [5]
    // d_in[5] is k == 64 (compile-time TOPK)

    float* out = (float*)d_out;                  // [512*CHW] mask ++ [512*5] logits

    float* scores = (float*)d_ws;                // CHW floats
    float* mask   = scores + CHW;                // CHW floats
    int*   topk   = (int*)(mask + CHW);          // 64 ints
    float* fmv    = (float*)(topk + TOPK);       // 64 floats

    mask_topk_kernel<<<1, 1024, 0, stream>>>(mlog, gum, scores, mask, topk, fmv);
    broadcast_mask_kernel<<<dim3(96 * 8), 256, 0, stream>>>(mask, out);
    logits_wmma_kernel<<<dim3(4), 256, 0, stream>>>(
        x, Wh, bh, topk, fmv, out + (size_t)BATCH * CHW);
}